// LightGCN_64819646431530
// MI455X (gfx1250) — compile-verified
//
#include <hip/hip_runtime.h>
#include <hip/hip_bf16.h>
#include <stdint.h>

#define EMB    64
#define CHUNK  1024   // edges staged to LDS per block (3 x 4KB LDS)

// ---------------------------------------------------------------- utilities
__global__ void zero_f32(float* __restrict__ p, int n) {
  int i = blockIdx.x * blockDim.x + threadIdx.x;
  int s = gridDim.x * blockDim.x;
  for (; i < n; i += s) p[i] = 0.0f;
}

// deg[col[e]] += 1  (float atomic: exact for counts < 2^24)
__global__ void degree_kernel(const int* __restrict__ col, float* __restrict__ deg, int E) {
  int i = blockIdx.x * blockDim.x + threadIdx.x;
  int s = gridDim.x * blockDim.x;
  for (; i < E; i += s) atomicAdd(&deg[col[i]], 1.0f);
}

// in-place: dis[i] = deg>0 ? rsqrt(deg) : 0   (deg>0 implies deg>=1 == max(deg,1))
__global__ void dis_kernel(float* __restrict__ deg, int N) {
  int i = blockIdx.x * blockDim.x + threadIdx.x;
  if (i < N) {
    float d = deg[i];
    deg[i] = (d > 0.0f) ? rsqrtf(d) : 0.0f;
  }
}

// norm[e] = dis[row[e]] * dis[col[e]]  (computed once, reused 3 layers)
__global__ void norm_kernel(const int* __restrict__ row, const int* __restrict__ col,
                            const float* __restrict__ dis, float* __restrict__ norm, int E) {
  int i = blockIdx.x * blockDim.x + threadIdx.x;
  int s = gridDim.x * blockDim.x;
  for (; i < E; i += s) norm[i] = dis[row[i]] * dis[col[i]];
}

// acc = total = concat(user_emb, movie_emb)
__global__ void init_kernel(const float* __restrict__ ue, const float* __restrict__ me,
                            float* __restrict__ acc, float* __restrict__ total,
                            int nU64, int nTot64) {
  int i = blockIdx.x * blockDim.x + threadIdx.x;
  int s = gridDim.x * blockDim.x;
  for (; i < nTot64; i += s) {
    float v = (i < nU64) ? ue[i] : me[i - nU64];
    acc[i] = v;
    total[i] = v;
  }
}

__global__ void add_kernel(float* __restrict__ out, const float* __restrict__ a, int n) {
  int i = blockIdx.x * blockDim.x + threadIdx.x;
  int s = gridDim.x * blockDim.x;
  for (; i < n; i += s) out[i] += a[i];
}

__global__ void final_kernel(float* __restrict__ out, const float* __restrict__ a,
                             float scale, int n) {
  int i = blockIdx.x * blockDim.x + threadIdx.x;
  int s = gridDim.x * blockDim.x;
  for (; i < n; i += s) out[i] = (out[i] + a[i]) * scale;
}

// ------------------------------------------------------------- SpMM scatter
// One block owns CHUNK consecutive edges. row/col/norm for the chunk are
// DMA'd into LDS with gfx1250 async global->LDS b128 loads (ASYNCcnt, then
// s_wait_asynccnt + barrier). Each half-wave (16 lanes, wave32) processes one
// edge: lane k gathers float4 k of the 64-float source row (256B coalesced),
// scales by the LDS-broadcast norm, and issues 4 global f32 atomic adds into
// the target row. All accumulators are L2-resident (192MB L2), so the layer
// is bounded by L2 atomic throughput, not HBM.
__global__ __launch_bounds__(256)
void scatter_kernel(const int* __restrict__ row, const int* __restrict__ col,
                    const float* __restrict__ norm,
                    const float* __restrict__ accIn, float* __restrict__ accOut,
                    int E) {
  __shared__ __align__(16) int   srow[CHUNK];
  __shared__ __align__(16) int   scol[CHUNK];
  __shared__ __align__(16) float snrm[CHUNK];

  int chunkStart = blockIdx.x * CHUNK;
  int count = E - chunkStart;
  if (count <= 0) return;
  if (count > CHUNK) count = CHUNK;
  int tid = threadIdx.x;

  // ---- async stage edge chunk into LDS (16B granules) -------------------
  int granules = count >> 2;                          // full 4-elem granules
  uint32_t ldsRow = (uint32_t)(uintptr_t)&srow[0];    // low 32 bits of generic
  uint32_t ldsCol = (uint32_t)(uintptr_t)&scol[0];    // shared ptr == LDS addr
  uint32_t ldsNrm = (uint32_t)(uintptr_t)&snrm[0];
  if (tid < granules) {
    uint32_t go = ((uint32_t)chunkStart + ((uint32_t)tid << 2)) * 4u;  // bytes
    uint32_t lo = (uint32_t)tid << 4;
    asm volatile("global_load_async_to_lds_b128 %0, %1, %2"
                 :: "v"(ldsRow + lo), "v"(go),
                    "s"((unsigned long long)(uintptr_t)row) : "memory");
    asm volatile("global_load_async_to_lds_b128 %0, %1, %2"
                 :: "v"(ldsCol + lo), "v"(go),
                    "s"((unsigned long long)(uintptr_t)col) : "memory");
    asm volatile("global_load_async_to_lds_b128 %0, %1, %2"
                 :: "v"(ldsNrm + lo), "v"(go),
                    "s"((unsigned long long)(uintptr_t)norm) : "memory");
  }
  if (tid == 0) {                                     // 0..3 tail edges
    for (int i = granules << 2; i < count; ++i) {
      srow[i] = row[chunkStart + i];
      scol[i] = col[chunkStart + i];
      snrm[i] = norm[chunkStart + i];
    }
  }
  asm volatile("s_wait_asynccnt 0" ::: "memory");
  __syncthreads();

  // ---- process: half-wave per edge --------------------------------------
  int sub   = tid & 15;   // which float4 of the 64-float row
  int eslot = tid >> 4;   // which edge slot (16 concurrent edges per pass)
  for (int i = eslot; i < count; i += 16) {
    int   r   = srow[i];
    int   c   = scol[i];
    float nrm = snrm[i];
    if (i + 16 < count) {  // prefetch next data-dependent gather row
      __builtin_prefetch(&accIn[(long long)srow[i + 16] * EMB + (sub << 2)], 0, 3);
    }
    const float4 v = *(const float4*)&accIn[(long long)r * EMB + (sub << 2)];
    float* dst = &accOut[(long long)c * EMB + (sub << 2)];
    atomicAdd(dst + 0, v.x * nrm);
    atomicAdd(dst + 1, v.y * nrm);
    atomicAdd(dst + 2, v.z * nrm);
    atomicAdd(dst + 3, v.w * nrm);
  }
}

// ------------------------------------------------------------------ launch
extern "C" void kernel_launch(void* const* d_in, const int* in_sizes, int n_in,
                              void* d_out, int out_size, void* d_ws, size_t ws_size,
                              hipStream_t stream) {
  const int*   edge = (const int*)d_in[0];
  const float* ue   = (const float*)d_in[2];
  const float* me   = (const float*)d_in[3];
  float*       out  = (float*)d_out;

  int E  = in_sizes[0] / 2;
  int nU = in_sizes[2] / EMB;
  int nM = in_sizes[3] / EMB;
  int N  = nU + nM;
  int NV = N * EMB;

  const int* row = edge;       // edge_index[0]
  const int* col = edge + E;   // edge_index[1]

  // workspace: dis[N] | norm[E] | accA[N*64] | accB[N*64]  (~97.4 MB)
  char*  ws  = (char*)d_ws;
  float* dis = (float*)ws;
  size_t off = (((size_t)N * 4) + 255) & ~(size_t)255;
  float* nrm = (float*)(ws + off);
  off += (((size_t)E * 4) + 255) & ~(size_t)255;
  float* accA = (float*)(ws + off);
  off += (((size_t)NV * 4) + 255) & ~(size_t)255;
  float* accB = (float*)(ws + off);

  const int tpb = 256;

  // degrees -> dis -> per-edge norm
  zero_f32<<<(N + tpb - 1) / tpb, tpb, 0, stream>>>(dis, N);
  degree_kernel<<<2048, tpb, 0, stream>>>(col, dis, E);
  dis_kernel<<<(N + tpb - 1) / tpb, tpb, 0, stream>>>(dis, N);
  norm_kernel<<<2048, tpb, 0, stream>>>(row, col, dis, nrm, E);

  // acc0 = total = x
  init_kernel<<<2048, tpb, 0, stream>>>(ue, me, accA, out, nU * EMB, NV);

  // 3 propagation layers (n_layers fixed by setup; host-unrolled for capture)
  int nBlocks = (E + CHUNK - 1) / CHUNK;
  float* cin  = accA;
  float* cout = accB;
  for (int layer = 0; layer < 3; ++layer) {
    zero_f32<<<2048, tpb, 0, stream>>>(cout, NV);
    scatter_kernel<<<nBlocks, tpb, 0, stream>>>(row, col, nrm, cin, cout, E);
    if (layer < 2)
      add_kernel<<<2048, tpb, 0, stream>>>(out, cout, NV);
    else
      final_kernel<<<2048, tpb, 0, stream>>>(out, cout, 0.25f, NV);
    float* t = cin; cin = cout; cout = t;
  }
}